// DRewGNNStage_33964601377216
// MI455X (gfx1250) — compile-verified
//
#include <hip/hip_runtime.h>

typedef __attribute__((ext_vector_type(2))) float v2f;
typedef __attribute__((ext_vector_type(8))) float v8f;

#define DIM 128
#define DIM4 (DIM / 4)

// ---------------------------------------------------------------------------
// init: agg buffers (2*N*D floats) = 0, degree buffers (2*N floats) = 1.0
// (the +1.0 is the GCN self-loop added to the segment-summed degree)
// ---------------------------------------------------------------------------
__global__ __launch_bounds__(256) void init_ws(float* __restrict__ agg, int aggCount,
                                               float* __restrict__ deg, int degCount) {
  int stride = gridDim.x * blockDim.x;
  for (int j = blockIdx.x * blockDim.x + threadIdx.x; j < aggCount; j += stride)
    agg[j] = 0.0f;
  for (int j = blockIdx.x * blockDim.x + threadIdx.x; j < degCount; j += stride)
    deg[j] = 1.0f;
}

// ---------------------------------------------------------------------------
// degree: deg_k[dst] += 1 for every edge with attr == k (k in {1,2})
// ---------------------------------------------------------------------------
__global__ __launch_bounds__(256) void degree_kernel(const int* __restrict__ dst,
                                                     const int* __restrict__ attr,
                                                     float* __restrict__ deg1,
                                                     float* __restrict__ deg2, int E) {
  int e = blockIdx.x * blockDim.x + threadIdx.x;
  if (e >= E) return;
  int a = attr[e];
  if (a == 1)      atomicAdd(&deg1[dst[e]], 1.0f);
  else if (a == 2) atomicAdd(&deg2[dst[e]], 1.0f);
}

// ---------------------------------------------------------------------------
// deg -> dinv = rsqrt(deg), in place (deg1/deg2 are contiguous: 2N floats)
// ---------------------------------------------------------------------------
__global__ __launch_bounds__(256) void to_dinv(float* __restrict__ deg, int n) {
  int i = blockIdx.x * blockDim.x + threadIdx.x;
  if (i < n) deg[i] = rsqrtf(deg[i]);
}

// ---------------------------------------------------------------------------
// GEMM via V_WMMA_F32_16X16X4_F32: h1 = x @ W1, h2 = x @ W2
// Block = 256 threads = 8 waves; block b handles rows [16b, 16b+16); wave w
// handles cols [16w, 16w+16) and accumulates both weight matrices so the A
// fragment (x tile) is reused twice per K-step.
//
// Fragment layouts per CDNA5 ISA 7.12.2 (wave32):
//   A 16x4 (MxK):  lanes 0-15 -> M=lane, V0=K0 V1=K1 ; lanes 16-31 -> V0=K2 V1=K3
//   B 4x16 (KxN):  rows striped across lanes per VGPR: V0 = {K0 | K2}, V1 = {K1 | K3}
//   C/D 16x16:     VGPR r: lanes 0-15 -> M=r, N=lane; lanes 16-31 -> M=r+8
// ---------------------------------------------------------------------------
__global__ __launch_bounds__(256) void gemm_wmma(const float* __restrict__ x,
                                                 const float* __restrict__ W1,
                                                 const float* __restrict__ W2,
                                                 float* __restrict__ h1,
                                                 float* __restrict__ h2) {
  const int wave = threadIdx.x >> 5;   // 0..7 -> column tile
  const int lane = threadIdx.x & 31;
  const int hi   = lane >> 4;          // lane-half (selects K pair)
  const int m    = lane & 15;
  const int row0 = blockIdx.x * 16;
  const int col0 = wave * 16;

  const float* xr = x + (size_t)(row0 + m) * DIM;

  v8f c1 = {};
  v8f c2 = {};
  for (int k = 0; k < DIM; k += 4) {
    const int ka = k + 2 * hi;
    v2f a;
    a.x = xr[ka + 0];
    a.y = xr[ka + 1];
    v2f b1f, b2f;
    b1f.x = W1[(size_t)(ka + 0) * DIM + col0 + m];
    b1f.y = W1[(size_t)(ka + 1) * DIM + col0 + m];
    b2f.x = W2[(size_t)(ka + 0) * DIM + col0 + m];
    b2f.y = W2[(size_t)(ka + 1) * DIM + col0 + m];
    c1 = __builtin_amdgcn_wmma_f32_16x16x4_f32(false, a, false, b1f, (short)0, c1,
                                               false, false);
    c2 = __builtin_amdgcn_wmma_f32_16x16x4_f32(false, a, false, b2f, (short)0, c2,
                                               false, false);
  }

#pragma unroll
  for (int r = 0; r < 8; ++r) {
    const size_t o = (size_t)(row0 + r + 8 * hi) * DIM + col0 + m;
    h1[o] = c1[r];
    h2[o] = c2[r];
  }
}

// ---------------------------------------------------------------------------
// scatter: one wave32 per edge. agg_k[dst] += h_k[src] * dinv_k[src]*dinv_k[dst]
// Each lane owns 4 consecutive columns (float4 gather + 4 f32 atomics).
// ---------------------------------------------------------------------------
__global__ __launch_bounds__(256) void scatter_kernel(const int* __restrict__ src,
                                                      const int* __restrict__ dst,
                                                      const int* __restrict__ attr,
                                                      const float* __restrict__ h1,
                                                      const float* __restrict__ h2,
                                                      const float* __restrict__ dinv1,
                                                      const float* __restrict__ dinv2,
                                                      float* __restrict__ agg1,
                                                      float* __restrict__ agg2, int E) {
  const int wave = threadIdx.x >> 5;
  const int lane = threadIdx.x & 31;
  const int e = blockIdx.x * 8 + wave;
  if (e >= E) return;

  const int a = attr[e];
  if (a == 0) return;
  const int s = src[e];
  const int d = dst[e];

  const float* h;
  const float* dinv;
  float* agg;
  if (a == 1) { h = h1; dinv = dinv1; agg = agg1; }
  else        { h = h2; dinv = dinv2; agg = agg2; }

  const float coef = dinv[s] * dinv[d];
  const float4 v = ((const float4*)(h + (size_t)s * DIM))[lane];
  float* ad = agg + (size_t)d * DIM + lane * 4;
  atomicAdd(ad + 0, v.x * coef);
  atomicAdd(ad + 1, v.y * coef);
  atomicAdd(ad + 2, v.z * coef);
  atomicAdd(ad + 3, v.w * coef);
}

// ---------------------------------------------------------------------------
// finalize: out = x + relu(a0*(agg1 + h1*dinv1^2 + b1) + a1*(agg2 + h2*dinv2^2 + b2))
// ---------------------------------------------------------------------------
__global__ __launch_bounds__(256) void finalize_kernel(const float* __restrict__ x,
                                                       const float* __restrict__ h1,
                                                       const float* __restrict__ h2,
                                                       const float* __restrict__ agg1,
                                                       const float* __restrict__ agg2,
                                                       const float* __restrict__ dinv1,
                                                       const float* __restrict__ dinv2,
                                                       const float* __restrict__ b1,
                                                       const float* __restrict__ b2,
                                                       const float* __restrict__ alpha,
                                                       float* __restrict__ out, int total4) {
  const int i = blockIdx.x * blockDim.x + threadIdx.x;
  if (i >= total4) return;

  const float al0 = alpha[0], al1 = alpha[1];
  const float mx = fmaxf(al0, al1);
  const float e0 = __expf(al0 - mx);
  const float e1 = __expf(al1 - mx);
  const float inv = 1.0f / (e0 + e1);
  const float a0 = e0 * inv;
  const float a1 = e1 * inv;

  const int node = i / DIM4;
  const int c4 = (i - node * DIM4) * 4;

  float s1 = dinv1[node]; s1 *= s1;   // 1/deg1
  float s2 = dinv2[node]; s2 *= s2;   // 1/deg2

  const float4 xv = ((const float4*)x)[i];
  const float4 v1 = ((const float4*)h1)[i];
  const float4 v2 = ((const float4*)h2)[i];
  const float4 g1 = ((const float4*)agg1)[i];
  const float4 g2 = ((const float4*)agg2)[i];

  float4 o;
  {
    float t;
    t = a0 * (g1.x + v1.x * s1 + b1[c4 + 0]) + a1 * (g2.x + v2.x * s2 + b2[c4 + 0]);
    o.x = xv.x + fmaxf(t, 0.0f);
    t = a0 * (g1.y + v1.y * s1 + b1[c4 + 1]) + a1 * (g2.y + v2.y * s2 + b2[c4 + 1]);
    o.y = xv.y + fmaxf(t, 0.0f);
    t = a0 * (g1.z + v1.z * s1 + b1[c4 + 2]) + a1 * (g2.z + v2.z * s2 + b2[c4 + 2]);
    o.z = xv.z + fmaxf(t, 0.0f);
    t = a0 * (g1.w + v1.w * s1 + b1[c4 + 3]) + a1 * (g2.w + v2.w * s2 + b2[c4 + 3]);
    o.w = xv.w + fmaxf(t, 0.0f);
  }
  ((float4*)out)[i] = o;
}

// ---------------------------------------------------------------------------
// launch
// inputs: 0:x [N,128] f32, 1:edge_index [2,E] i32, 2:edge_attr [E] i32,
//         3:W1 [128,128] f32, 4:b1 [128] f32, 5:W2 f32, 6:b2 f32, 7:alpha [2] f32
// ---------------------------------------------------------------------------
extern "C" void kernel_launch(void* const* d_in, const int* in_sizes, int n_in,
                              void* d_out, int out_size, void* d_ws, size_t ws_size,
                              hipStream_t stream) {
  const float* x     = (const float*)d_in[0];
  const int*   eidx  = (const int*)d_in[1];
  const int*   eattr = (const int*)d_in[2];
  const float* W1    = (const float*)d_in[3];
  const float* b1    = (const float*)d_in[4];
  const float* W2    = (const float*)d_in[5];
  const float* b2    = (const float*)d_in[6];
  const float* alpha = (const float*)d_in[7];
  float* out = (float*)d_out;

  const int N = in_sizes[0] / DIM;   // 50000
  const int E = in_sizes[2];         // 640000
  const int* esrc = eidx;
  const int* edst = eidx + E;

  // workspace carve: h1 | h2 | agg1 | agg2 | dinv1 | dinv2
  float* h1    = (float*)d_ws;
  float* h2    = h1 + (size_t)N * DIM;
  float* agg1  = h2 + (size_t)N * DIM;
  float* agg2  = agg1 + (size_t)N * DIM;
  float* dinv1 = agg2 + (size_t)N * DIM;
  float* dinv2 = dinv1 + N;

  // 1) init agg=0, deg=1
  init_ws<<<1024, 256, 0, stream>>>(agg1, 2 * N * DIM, dinv1, 2 * N);

  // 2) degree accumulation
  degree_kernel<<<(E + 255) / 256, 256, 0, stream>>>(edst, eattr, dinv1, dinv2, E);

  // 3) GEMMs via WMMA (independent of degree; stream order is fine)
  gemm_wmma<<<N / 16, 256, 0, stream>>>(x, W1, W2, h1, h2);

  // 4) deg -> rsqrt(deg)
  to_dinv<<<(2 * N + 255) / 256, 256, 0, stream>>>(dinv1, 2 * N);

  // 5) edge scatter (one wave per edge)
  scatter_kernel<<<(E + 7) / 8, 256, 0, stream>>>(esrc, edst, eattr, h1, h2,
                                                  dinv1, dinv2, agg1, agg2, E);

  // 6) finalize
  const int total4 = N * DIM4;
  finalize_kernel<<<(total4 + 255) / 256, 256, 0, stream>>>(
      x, h1, h2, agg1, agg2, dinv1, dinv2, b1, b2, alpha, out, total4);
}